// ConvBlock_20194936225909
// MI455X (gfx1250) — compile-verified
//
#include <hip/hip_runtime.h>

typedef __attribute__((ext_vector_type(2))) float v2f;
typedef __attribute__((ext_vector_type(8))) float v8f;

#define D 64

// ---------------------------------------------------------------------------
// Zero a float buffer (float4-wide).
// ---------------------------------------------------------------------------
__global__ __launch_bounds__(256) void zero_kernel(float4* __restrict__ p, long long n4) {
    long long i = (long long)blockIdx.x * blockDim.x + threadIdx.x;
    if (i < n4) p[i] = make_float4(0.f, 0.f, 0.f, 0.f);
}

// ---------------------------------------------------------------------------
// Edge scatter: agg[dst] += x[src] * w.  16 lanes per edge, float4 per lane.
// Gathers hit L2 (x is 25.6MB << 192MB L2); atomics resolve in L2.
// ---------------------------------------------------------------------------
__global__ __launch_bounds__(256) void scatter_edges(
    const float* __restrict__ feat,
    const long long* __restrict__ srcIdx,
    const long long* __restrict__ dstIdx,
    const float* __restrict__ ew,
    float* __restrict__ agg,
    long long E)
{
    long long gid = (long long)blockIdx.x * blockDim.x + threadIdx.x;
    long long e = gid >> 4;          // 16 threads per edge
    int c = (int)(gid & 15);         // 4-float chunk within the 64-wide row
    if (e >= E) return;

    long long s = srcIdx[e];
    long long d = dstIdx[e];
    float w = ew[e];

    const float4 v = *(const float4*)(feat + s * (long long)D + c * 4);
    float* p = agg + d * (long long)D + c * 4;
    atomicAdd(p + 0, v.x * w);
    atomicAdd(p + 1, v.y * w);
    atomicAdd(p + 2, v.z * w);
    atomicAdd(p + 3, v.w * w);
}

// ---------------------------------------------------------------------------
// Fused dual-GEMM + bias + optional ReLU, f32 WMMA 16x16x4:
//   out = act( agg @ Wrel^T + brel + xin @ Wroot^T )
// Block = 128 threads = 4 waves; block handles a 16-row strip, each wave a
// 16-column tile.  K accumulated in 16 steps of 4 per input matrix.
//
// A layout (32-bit 16x4): lane L supplies M = L&15; VGPR0/1 = K {k0,k0+1}
// for lanes 0-15, K {k0+2,k0+3} for lanes 16-31  -> contiguous float2 load.
// B layout mirrors A with N = L&15 (B[k][n] = W[n][k], W row-major).
// C/D layout: VGPR v -> M = v + 8*(lane>=16), N = lane&15.
// ---------------------------------------------------------------------------
__global__ __launch_bounds__(128) void gemm_fused(
    const float* __restrict__ agg,
    const float* __restrict__ xin,
    const float* __restrict__ Wrel,
    const float* __restrict__ brel,
    const float* __restrict__ Wroot,
    float* __restrict__ out,
    int do_relu)
{
    const int lane = threadIdx.x & 31;
    const int wave = threadIdx.x >> 5;       // column tile 0..3
    const int rowTile = blockIdx.x;          // 16-row strip

    const int mn    = lane & 15;             // M for A loads, N for B loads
    const int khalf = (lane >> 4) * 2;       // lanes 16-31 handle K+2,K+3

    const long long arow = (long long)(rowTile * 16 + mn) * D;
    const long long brow = (long long)(wave * 16 + mn) * D;   // W row = output col

    v8f acc = {};

    #pragma unroll
    for (int k0 = 0; k0 < D; k0 += 4) {
        v2f a = *(const v2f*)(agg + arow + k0 + khalf);
        v2f b = *(const v2f*)(Wrel + brow + k0 + khalf);
        acc = __builtin_amdgcn_wmma_f32_16x16x4_f32(
            /*neg_a=*/false, a, /*neg_b=*/false, b,
            /*c_mod=*/(short)0, acc, /*reuse_a=*/false, /*reuse_b=*/false);
    }
    #pragma unroll
    for (int k0 = 0; k0 < D; k0 += 4) {
        v2f a = *(const v2f*)(xin + arow + k0 + khalf);
        v2f b = *(const v2f*)(Wroot + brow + k0 + khalf);
        acc = __builtin_amdgcn_wmma_f32_16x16x4_f32(
            false, a, false, b, (short)0, acc, false, false);
    }

    // Store C: VGPR v -> row rowTile*16 + v + 8*(lane>=16), col wave*16 + (lane&15)
    const int colg = wave * 16 + mn;
    const float bias = brel[colg];
    const int moff = (lane >> 4) * 8;

    #pragma unroll
    for (int v = 0; v < 8; ++v) {
        const int mg = rowTile * 16 + moff + v;
        float val = acc[v] + bias;
        if (do_relu) val = fmaxf(val, 0.f);
        out[(long long)mg * D + colg] = val;
    }
}

// ---------------------------------------------------------------------------
// Host launcher.  Inputs (setup_inputs order):
//   0 x[N,64] f32, 1 edge_index[2,E] i64, 2 edge_weight[E] f32,
//   3 W_rel1, 4 b_rel1, 5 W_root1, 6 W_rel2, 7 b_rel2, 8 W_root2
// ---------------------------------------------------------------------------
extern "C" void kernel_launch(void* const* d_in, const int* in_sizes, int n_in,
                              void* d_out, int out_size, void* d_ws, size_t ws_size,
                              hipStream_t stream) {
    const float*     x      = (const float*)d_in[0];
    const long long* ei     = (const long long*)d_in[1];
    const float*     ew     = (const float*)d_in[2];
    const float*     Wrel1  = (const float*)d_in[3];
    const float*     brel1  = (const float*)d_in[4];
    const float*     Wroot1 = (const float*)d_in[5];
    const float*     Wrel2  = (const float*)d_in[6];
    const float*     brel2  = (const float*)d_in[7];
    const float*     Wroot2 = (const float*)d_in[8];
    float* out = (float*)d_out;

    const int       N = in_sizes[0] / D;            // 100000
    const long long E = (long long)in_sizes[1] / 2; // 1600000
    const long long* srcI = ei;
    const long long* dstI = ei + E;

    float* agg = (float*)d_ws;                      // N*64 f32
    float* h1  = agg + (long long)N * D;            // N*64 f32

    const long long n4 = (long long)N * D / 4;
    const unsigned zeroBlocks    = (unsigned)((n4 + 255) / 256);
    const unsigned scatterBlocks = (unsigned)((E * 16 + 255) / 256);
    const unsigned gemmBlocks    = (unsigned)(N / 16);  // N divisible by 16

    // ---- layer 1 ----
    zero_kernel<<<zeroBlocks, 256, 0, stream>>>((float4*)agg, n4);
    scatter_edges<<<scatterBlocks, 256, 0, stream>>>(x, srcI, dstI, ew, agg, E);
    gemm_fused<<<gemmBlocks, 128, 0, stream>>>(agg, x, Wrel1, brel1, Wroot1, h1, /*relu=*/1);

    // ---- layer 2 ----
    zero_kernel<<<zeroBlocks, 256, 0, stream>>>((float4*)agg, n4);
    scatter_edges<<<scatterBlocks, 256, 0, stream>>>(h1, srcI, dstI, ew, agg, E);
    gemm_fused<<<gemmBlocks, 128, 0, stream>>>(agg, h1, Wrel2, brel2, Wroot2, out, /*relu=*/0);
}